// TopicAttentiveEmbedding_30296699306346
// MI455X (gfx1250) — compile-verified
//
#include <hip/hip_runtime.h>

// Problem sizes (fixed by the reference).
#define B_ 64
#define S_ 4096
#define D_ 512

typedef __attribute__((ext_vector_type(2))) float v2f;
typedef __attribute__((ext_vector_type(8))) float v8f;

// ---------------------------------------------------------------------------
// Kernel 1: t = topic @ W   (64x512 = [64x512] * [512x512]), fp32 WMMA.
// One wave computes one 16x16 tile of t; K accumulated in 4-wide steps via
// V_WMMA_F32_16X16X4_F32 (128 WMMA issues per wave).
// grid = (D/16, B/16) = (32, 4), block = 32 (one wave32, EXEC all ones).
// ---------------------------------------------------------------------------
__global__ __launch_bounds__(32)
void topic_w_wmma_kernel(const float* __restrict__ topic,
                         const float* __restrict__ W,
                         float* __restrict__ t_out)
{
    const int lane = threadIdx.x & 31;
    const int half = lane >> 4;      // 0: lanes 0-15, 1: lanes 16-31
    const int lr   = lane & 15;
    const int n0   = blockIdx.x * 16;  // column tile (over D)
    const int m0   = blockIdx.y * 16;  // row tile (over B)

    v8f c = {};
    const float* arow = topic + (size_t)(m0 + lr) * D_;

    #pragma unroll 4
    for (int k0 = 0; k0 < D_; k0 += 4) {
        // A fragment (16x4, M x K): per lane two consecutive K values;
        // lanes 0-15 hold K={k0,k0+1}, lanes 16-31 hold K={k0+2,k0+3}.
        v2f a = *(const v2f*)&arow[k0 + 2 * half];
        // B fragment (4x16, K x N): row K striped across half-waves, N = lr.
        v2f b;
        b.x = W[(size_t)(k0 + 2 * half)     * D_ + n0 + lr];
        b.y = W[(size_t)(k0 + 2 * half + 1) * D_ + n0 + lr];
        // 8 args: (neg_a, A, neg_b, B, c_mod, C, reuse_a, reuse_b)
        c = __builtin_amdgcn_wmma_f32_16x16x4_f32(false, a, false, b,
                                                  (short)0, c, false, false);
    }

    // D tile: VGPR r holds row (r + 8*half), column lr.
    #pragma unroll
    for (int r = 0; r < 8; ++r) {
        t_out[(size_t)(m0 + r + 8 * half) * D_ + n0 + lr] = c[r];
    }
}

// ---------------------------------------------------------------------------
// Kernel 2: scores[b,s] = t[b] . seq[b,s] + bias.  Streams the 512 MB tensor
// once at full coalescing: one wave per token, lane l reads float4s at
// d = 4*l + 128*k (k=0..3) -> each k step is a dense 512B burst per wave.
// t[b] staged via LDS into 16 VGPRs per lane. Memory-bound by design.
// grid = (S/512, B), block = 256 (8 waves); each wave loops 64 tokens.
// ---------------------------------------------------------------------------
#define TOKENS_PER_BLOCK 512

__global__ __launch_bounds__(256)
void scores_kernel(const float* __restrict__ seq,
                   const float* __restrict__ t,
                   const float* __restrict__ bias,
                   float* __restrict__ scores)
{
    __shared__ float t_lds[D_];
    const int tid = threadIdx.x;
    const int b   = blockIdx.y;

    for (int i = tid; i < D_; i += 256) t_lds[i] = t[(size_t)b * D_ + i];
    __syncthreads();

    const float bias0 = bias[0];
    const int wave = tid >> 5;
    const int lane = tid & 31;

    // Pull this lane's slice of t[b] into registers once.
    const float4* tv4 = (const float4*)t_lds;
    float4 tw[4];
    #pragma unroll
    for (int k = 0; k < 4; ++k) tw[k] = tv4[lane + 32 * k];

    const float* seqb  = seq + (size_t)b * S_ * D_;
    const int     s_end = (blockIdx.x + 1) * TOKENS_PER_BLOCK;

    for (int s = blockIdx.x * TOKENS_PER_BLOCK + wave; s < s_end; s += 8) {
        const float4* row = (const float4*)(seqb + (size_t)s * D_);
        // Speculative prefetch of this wave's next token (s+8) -> global_prefetch_b8.
        __builtin_prefetch((const void*)(row + 8 * (D_ / 4) + lane), 0, 1);

        float acc = 0.f;
        #pragma unroll
        for (int k = 0; k < 4; ++k) {
            float4 v = row[lane + 32 * k];   // global_load_b128, coalesced
            float4 w = tw[k];
            acc += v.x * w.x + v.y * w.y + v.z * w.z + v.w * w.w;
        }
        // wave32 all-reduce
        #pragma unroll
        for (int off = 16; off >= 1; off >>= 1)
            acc += __shfl_xor(acc, off, 32);
        if (lane == 0) scores[(size_t)b * S_ + s] = acc + bias0;
    }
}

// ---------------------------------------------------------------------------
// Kernel 3: row-wise softmax over S=4096.  One block per batch row.
// block = 256 threads, 16 elements per thread held in registers.
// ---------------------------------------------------------------------------
__global__ __launch_bounds__(256)
void softmax_kernel(const float* __restrict__ scores, float* __restrict__ out)
{
    __shared__ float sred[8];
    const int b    = blockIdx.x;
    const int tid  = threadIdx.x;
    const int wave = tid >> 5;
    const int lane = tid & 31;

    const float* row  = scores + (size_t)b * S_;
    float*       orow = out    + (size_t)b * S_;

    float loc[16];
    float m = -3.402823466e+38f;
    #pragma unroll
    for (int i = 0; i < 16; ++i) {
        loc[i] = row[tid + 256 * i];
        m = fmaxf(m, loc[i]);
    }
    #pragma unroll
    for (int off = 16; off >= 1; off >>= 1)
        m = fmaxf(m, __shfl_xor(m, off, 32));
    if (lane == 0) sred[wave] = m;
    __syncthreads();
    float M = sred[0];
    #pragma unroll
    for (int w = 1; w < 8; ++w) M = fmaxf(M, sred[w]);
    __syncthreads();   // sred reused below

    float sum = 0.f;
    #pragma unroll
    for (int i = 0; i < 16; ++i) {
        loc[i] = __expf(loc[i] - M);
        sum += loc[i];
    }
    #pragma unroll
    for (int off = 16; off >= 1; off >>= 1)
        sum += __shfl_xor(sum, off, 32);
    if (lane == 0) sred[wave] = sum;
    __syncthreads();
    float T = 0.f;
    #pragma unroll
    for (int w = 0; w < 8; ++w) T += sred[w];
    const float inv = 1.0f / T;

    #pragma unroll
    for (int i = 0; i < 16; ++i)
        orow[tid + 256 * i] = loc[i] * inv;
}

// ---------------------------------------------------------------------------
// Launch: inputs in setup_inputs() order:
//   d_in[0] topic_embedding [B,D] f32
//   d_in[1] sequence_embedding [B,S,D] f32
//   d_in[2] W [D,D] f32
//   d_in[3] bias [1] f32
// d_out: beta [B,S] f32.  Workspace: t (B*D f32) | scores (B*S f32) ~1.13 MB.
// ---------------------------------------------------------------------------
extern "C" void kernel_launch(void* const* d_in, const int* in_sizes, int n_in,
                              void* d_out, int out_size, void* d_ws, size_t ws_size,
                              hipStream_t stream)
{
    const float* topic = (const float*)d_in[0];
    const float* seq   = (const float*)d_in[1];
    const float* W     = (const float*)d_in[2];
    const float* bias  = (const float*)d_in[3];
    float*       out   = (float*)d_out;

    float* t_ws     = (float*)d_ws;            // B*D floats
    float* score_ws = t_ws + (size_t)B_ * D_;  // B*S floats

    dim3 g1(D_ / 16, B_ / 16);                       // (32, 4)
    topic_w_wmma_kernel<<<g1, 32, 0, stream>>>(topic, W, t_ws);

    dim3 g2(S_ / TOKENS_PER_BLOCK, B_);              // (8, 64)
    scores_kernel<<<g2, 256, 0, stream>>>(seq, t_ws, bias, score_ws);

    softmax_kernel<<<B_, 256, 0, stream>>>(score_ws, out);
}